// Topk_CE_Dice_Loss_35107062677770
// MI455X (gfx1250) — compile-verified
//
#include <hip/hip_runtime.h>
#include <hip/hip_bf16.h>

typedef __attribute__((ext_vector_type(2))) float v2f;
typedef __attribute__((ext_vector_type(8))) float v8f;

#define QN 16
#define PIX 262144            // 512*512
#define NBLK 128              // blocks per image in pass 1
#define BN 8
#define EPS_F 1e-6f
#define SMOOTH_F 1e-4f

// workspace layout (floats)
#define PS_OFF 0
#define IN_OFF (BN * NBLK * QN)          // 16384
#define TS_OFF (2 * BN * NBLK * QN)      // 32768
#define CE_OFF (3 * BN * NBLK * QN)      // 49152
#define VC_OFF (3 * BN * NBLK * QN + BN * NBLK) // 50176
// total: 51200 floats = 204800 bytes

__global__ __launch_bounds__(256) void ce_dice_pass1(
    const float* __restrict__ inputs,
    const int* __restrict__ targets,
    const int* __restrict__ objnum,
    float* __restrict__ ws)
{
  const int tid = threadIdx.x;
  const int blk = blockIdx.x;      // 0..127
  const int b   = blockIdx.y;      // 0..7
  const int numb = objnum[b];
  const float* ib = inputs + (size_t)b * QN * PIX;
  const int* tb   = targets + (size_t)b * PIX;

  float psumA[QN], interA[QN], tsumA[QN];
#pragma unroll
  for (int q = 0; q < QN; ++q) { psumA[q] = 0.f; interA[q] = 0.f; tsumA[q] = 0.f; }
  float ceA = 0.f, vcA = 0.f;

#pragma unroll
  for (int it = 0; it < 4; ++it) {
    const int p0 = blk * 2048 + it * 512 + tid * 2;  // 2 consecutive pixels

    float2 xa[QN];
#pragma unroll
    for (int q = 0; q < QN; ++q)
      xa[q] = *reinterpret_cast<const float2*>(ib + (size_t)q * PIX + p0);

    float m0 = xa[0].x, m1 = xa[0].y;
#pragma unroll
    for (int q = 1; q < QN; ++q) { m0 = fmaxf(m0, xa[q].x); m1 = fmaxf(m1, xa[q].y); }

    float d0 = 0.f, d1 = 0.f;
#pragma unroll
    for (int q = 0; q < QN; ++q) {
      float e0 = __expf(xa[q].x - m0);
      float e1 = __expf(xa[q].y - m1);
      xa[q].x = e0; xa[q].y = e1;
      d0 += e0; d1 += e1;
    }
    const float i0 = __builtin_amdgcn_rcpf(d0);
    const float i1 = __builtin_amdgcn_rcpf(d1);

    const int t0 = tb[p0];
    const int t1 = tb[p0 + 1];
    float pt0 = 0.f, pt1 = 0.f;
#pragma unroll
    for (int q = 0; q < QN; ++q) {
      const float pa = xa[q].x * i0;
      const float pb = xa[q].y * i1;
      psumA[q] += pa + pb;
      const float f0 = (t0 == q) ? 1.f : 0.f;
      const float f1 = (t1 == q) ? 1.f : 0.f;
      interA[q] = fmaf(f0, pa, fmaf(f1, pb, interA[q]));
      tsumA[q] += f0 + f1;
      pt0 = fmaf(f0, pa, pt0);
      pt1 = fmaf(f1, pb, pt1);
    }
    // cross entropy with validity mask
    {
      const bool v0 = (t0 > 0) && (t0 <= numb);
      const float c0 = fminf(fmaxf(-__logf(pt0 + EPS_F), EPS_F), 20.f);
      ceA += v0 ? c0 : 0.f;  vcA += v0 ? 1.f : 0.f;
      const bool v1 = (t1 > 0) && (t1 <= numb);
      const float c1 = fminf(fmaxf(-__logf(pt1 + EPS_F), EPS_F), 20.f);
      ceA += v1 ? c1 : 0.f;  vcA += v1 ? 1.f : 0.f;
    }
  }

  // deterministic block reduction: 256 threads x 50 values via LDS (stride 51)
  __shared__ float buf[256 * 51];
  float* row = buf + tid * 51;
#pragma unroll
  for (int q = 0; q < QN; ++q) {
    row[q]      = psumA[q];
    row[16 + q] = interA[q];
    row[32 + q] = tsumA[q];
  }
  row[48] = ceA;
  row[49] = vcA;
  __syncthreads();

  if (tid < 50) {
    float s = 0.f;
    for (int i = 0; i < 256; ++i) s += buf[i * 51 + tid];
    const int bb = b * NBLK + blk;
    if (tid < 16)       ws[PS_OFF + bb * QN + tid]        = s;
    else if (tid < 32)  ws[IN_OFF + bb * QN + (tid - 16)] = s;
    else if (tid < 48)  ws[TS_OFF + bb * QN + (tid - 32)] = s;
    else if (tid == 48) ws[CE_OFF + bb] = s;
    else                ws[VC_OFF + bb] = s;
  }
}

__global__ __launch_bounds__(256) void ce_dice_pass2(
    const float* __restrict__ ws,
    const int* __restrict__ objnum,
    float* __restrict__ out)
{
  __shared__ float psumT[BN][QN];
  __shared__ float interT[BN][QN];
  __shared__ float tsumT[BN][QN];
  __shared__ float cbuf[256];
  __shared__ float vbuf[256];

  const int tid = threadIdx.x;

  // ce / valid-count partials over 1024 block entries
  float cs = 0.f, vs = 0.f;
  for (int i = tid; i < BN * NBLK; i += 256) {
    cs += ws[CE_OFF + i];
    vs += ws[VC_OFF + i];
  }
  cbuf[tid] = cs;
  vbuf[tid] = vs;

  // Each wave (b = wave id) reduces its 16(q) x 128(blk) partial matrices
  // for psum / inter / tsum with chained V_WMMA_F32_16X16X4_F32, B = ones:
  // D[m,n] = sum_k A[m,k] + C[m,n]  -> 32 WMMAs sum 128 blocks, pure f32.
  const int lane = tid & 31;
  const int wv   = tid >> 5;       // b = 0..7
  const int q    = lane & 15;      // A-matrix row M
  const int half = lane >> 4;      // lanes 16-31 carry K=2,3
  v2f bones; bones.x = 1.f; bones.y = 1.f;

#pragma unroll
  for (int s = 0; s < 3; ++s) {
    const int roff = s * BN * NBLK * QN;
    v8f acc;
#pragma unroll
    for (int r = 0; r < 8; ++r) acc[r] = 0.f;

    for (int base = 0; base < NBLK; base += 4) {
      const int blk = base + half * 2;
      const int idx = roff + (wv * NBLK + blk) * QN + q;
      v2f a;
      a.x = ws[idx];        // A[q][k]   (k = half*2)
      a.y = ws[idx + QN];   // A[q][k+1] (next block, same q)
      acc = __builtin_amdgcn_wmma_f32_16x16x4_f32(
          false, a, false, bones, (short)0, acc, false, false);
    }

    float* dst = (s == 0) ? &psumT[wv][0] : ((s == 1) ? &interT[wv][0] : &tsumT[wv][0]);
    // D column n=0: lane 0 holds M=0..7 in acc[0..7]; lane 16 holds M=8..15
    if (lane == 0) {
#pragma unroll
      for (int r = 0; r < 8; ++r) dst[r] = acc[r];
    }
    if (lane == 16) {
#pragma unroll
      for (int r = 0; r < 8; ++r) dst[8 + r] = acc[r];
    }
  }
  __syncthreads();

  if (tid == 0) {
    float ct = 0.f, vt = 0.f;
    for (int i = 0; i < 256; ++i) { ct += cbuf[i]; vt += vbuf[i]; }
    const float ce_loss = ct / (vt + 1.f);

    float dice = 0.f;
    for (int b = 0; b < BN; ++b) {
      const int num = objnum[b];
      float db = 0.f;
      for (int qq = 1; qq < QN; ++qq) {
        if (qq <= num) {
          db += 1.f - (2.f * interT[b][qq] + SMOOTH_F) /
                       (psumT[b][qq] + tsumT[b][qq] + SMOOTH_F);
        }
      }
      dice += db / (float)num;
    }
    dice *= (1.f / (float)BN);
    out[0] = ce_loss + dice;
  }
}

extern "C" void kernel_launch(void* const* d_in, const int* in_sizes, int n_in,
                              void* d_out, int out_size, void* d_ws, size_t ws_size,
                              hipStream_t stream) {
  (void)in_sizes; (void)n_in; (void)out_size; (void)ws_size;
  const float* inputs = (const float*)d_in[0];
  const int* targets  = (const int*)d_in[1];
  const int* objnum   = (const int*)d_in[2];
  float* ws  = (float*)d_ws;
  float* out = (float*)d_out;

  dim3 g1(NBLK, BN);
  ce_dice_pass1<<<g1, dim3(256), 0, stream>>>(inputs, targets, objnum, ws);
  ce_dice_pass2<<<dim3(1), dim3(256), 0, stream>>>(ws, objnum, out);
}